// Lstm_90323162235334
// MI455X (gfx1250) — compile-verified
//
#include <hip/hip_runtime.h>
#include <cstddef>

typedef __attribute__((ext_vector_type(2))) float v2f;
typedef __attribute__((ext_vector_type(8))) float v8f;

namespace {
constexpr int B_    = 256;
constexpr int T_    = 2048;
constexpr int FEAT_ = 8;
constexpr int D_    = 43;
constexpr int G_    = 4 * D_;     // 172 gate outputs (i,f,g,o)
constexpr int GP_   = 176;        // padded to 11 * 16
constexpr int NW_   = GP_ / 16;   // 11 waves per block
constexpr int BLK_  = NW_ * 32;   // 352 threads (wave32)
constexpr int MT_   = 16;         // batch rows per workgroup (one WMMA M tile)
constexpr int H1_   = 30;
constexpr int H2_   = 20;
constexpr int IN2_  = 2 * D_;     // 86
}

// Bidirectional LSTM layer: each block handles MT_ batch rows for one direction,
// iterating all T_ steps with h/c/weights resident in LDS.
// z = [in_t | h] @ [W_ih | W_hh]^T + (b_ih + b_hh), computed with V_WMMA_F32_16X16X4_F32.
template<int INLEN, int INPAD, int KP, int KW, bool FUSE_PROJ>
__global__ void lstm_bidir_kernel(
    const float* __restrict__ in,      // FUSE_PROJ ? x [B,T,FEAT] : h_prev [B,T,2D]
    const float* __restrict__ w_proj, const float* __restrict__ b_proj,
    const float* __restrict__ w_ih_f, const float* __restrict__ w_hh_f,
    const float* __restrict__ b_ih_f, const float* __restrict__ b_hh_f,
    const float* __restrict__ w_ih_b, const float* __restrict__ w_hh_b,
    const float* __restrict__ b_ih_b, const float* __restrict__ b_hh_b,
    float* __restrict__ hout)          // [B,T,2D], fwd in cols [0,D), bwd in [D,2D)
{
  extern __shared__ float smem[];
  float* wt   = smem;                    // [GP_][KW]  combined weights, gate-major
  float* bias = wt + GP_ * KW;           // [GP_]      b_ih + b_hh
  float* zbuf = bias + GP_;              // [MT_][GP_] pre-activation gates
  float* abuf = zbuf + MT_ * GP_;        // [MT_][KP]  [in_t | h] A-matrix rows
  float* cbuf = abuf + MT_ * KP;         // [MT_][D_+1] cell state
  float* wprj = cbuf + MT_ * (D_ + 1);   // [D_][FEAT_] (layer-1 fused projection)
  float* bprj = wprj + D_ * FEAT_;       // [D_]

  const int tid = threadIdx.x;
  const int dir = blockIdx.x & 1;
  const int b0  = (blockIdx.x >> 1) * MT_;

  const float* w_ih = dir ? w_ih_b : w_ih_f;
  const float* w_hh = dir ? w_hh_b : w_hh_f;
  const float* b_ih = dir ? b_ih_b : b_ih_f;
  const float* b_hh = dir ? b_hh_b : b_hh_f;

  // ---- init LDS: weights (zero-padded), bias, h=c=0 ----
  for (int idx = tid; idx < GP_ * KW; idx += BLK_) {
    int g = idx / KW, k = idx - g * KW;
    float v = 0.0f;
    if (g < G_) {
      if (k < INLEN)                          v = w_ih[g * INLEN + k];
      else if (k >= INPAD && k < INPAD + D_)  v = w_hh[g * D_ + (k - INPAD)];
    }
    wt[idx] = v;
  }
  for (int idx = tid; idx < GP_; idx += BLK_)
    bias[idx] = (idx < G_) ? (b_ih[idx] + b_hh[idx]) : 0.0f;
  for (int idx = tid; idx < MT_ * KP; idx += BLK_)       abuf[idx] = 0.0f;
  for (int idx = tid; idx < MT_ * (D_ + 1); idx += BLK_) cbuf[idx] = 0.0f;
  if (FUSE_PROJ) {
    for (int idx = tid; idx < D_ * FEAT_; idx += BLK_) wprj[idx] = w_proj[idx];
    for (int idx = tid; idx < D_; idx += BLK_)         bprj[idx] = b_proj[idx];
  }
  __syncthreads();

  const int lane = tid & 31;
  const int wv   = tid >> 5;
  const int mrow = lane & 15;           // A-matrix row / C-matrix N helper
  const int khi  = (lane >> 4) * 2;     // K sub-offset: 0 for lanes 0-15, 2 for 16-31
  const int col  = wv * 16 + mrow;      // gate column owned by this lane's tile

  for (int t = 0; t < T_; ++t) {
    const int tt = dir ? (T_ - 1 - t) : t;

    // ---- stage in_t into abuf[:, 0:INLEN] ----
    if (FUSE_PROJ) {
      for (int idx = tid; idx < MT_ * D_; idx += BLK_) {
        int r = idx / D_, j = idx - r * D_;
        const float* xr = in + ((size_t)(b0 + r) * T_ + tt) * FEAT_;
        float acc = bprj[j];
        #pragma unroll
        for (int k = 0; k < FEAT_; ++k) acc = fmaf(xr[k], wprj[j * FEAT_ + k], acc);
        abuf[r * KP + j] = acc;
      }
      if (tid < MT_ && t + 1 < T_) {
        const int tn = dir ? (T_ - 2 - t) : (t + 1);
        __builtin_prefetch(in + ((size_t)(b0 + tid) * T_ + tn) * FEAT_, 0, 0);
      }
    } else {
      for (int idx = tid; idx < MT_ * INLEN; idx += BLK_) {
        int r = idx / INLEN, j = idx - r * INLEN;
        abuf[r * KP + j] = in[((size_t)(b0 + r) * T_ + tt) * INLEN + j];
      }
      if (tid < MT_ && t + 1 < T_) {
        const int tn = dir ? (T_ - 2 - t) : (t + 1);
        __builtin_prefetch(in + ((size_t)(b0 + tid) * T_ + tn) * INLEN, 0, 0);
      }
    }
    __syncthreads();

    // ---- z tile: each wave computes one 16(M) x 16(N) gate tile over K=KP ----
    v8f acc0 = {};
    v8f acc1 = {};
    const float* arow = abuf + mrow * KP;   // A: lanes 0-15 row M, VGPR0/1 = K+khi,+khi+1
    const float* bcol = wt + col * KW;      // B: mirror layout, gate-major storage
    #pragma unroll
    for (int kk = 0; kk < KP / 4; kk += 2) {
      const int k0 = kk * 4 + khi;
      v2f a0 = *(const v2f*)(arow + k0);
      v2f w0 = *(const v2f*)(bcol + k0);
      acc0 = __builtin_amdgcn_wmma_f32_16x16x4_f32(false, a0, false, w0,
                                                   (short)0, acc0, false, false);
      const int k1 = k0 + 4;
      v2f a1v = *(const v2f*)(arow + k1);
      v2f w1v = *(const v2f*)(bcol + k1);
      acc1 = __builtin_amdgcn_wmma_f32_16x16x4_f32(false, a1v, false, w1v,
                                                   (short)0, acc1, false, false);
    }
    {
      // C/D layout: VGPR r -> M = r (+8 for upper half-wave), N = col
      const int rbase = (lane >> 4) * 8;
      const float bb = bias[col];
      #pragma unroll
      for (int r = 0; r < 8; ++r)
        zbuf[(rbase + r) * GP_ + col] = acc0[r] + acc1[r] + bb;
    }
    __syncthreads();

    // ---- LSTM cell update (i,f,g,o split along the 4*D gate axis) ----
    for (int idx = tid; idx < MT_ * D_; idx += BLK_) {
      int r = idx / D_, j = idx - r * D_;
      const float* zr = zbuf + r * GP_;
      float zi = zr[j], zf = zr[D_ + j], zg = zr[2 * D_ + j], zo = zr[3 * D_ + j];
      float gi = 1.0f / (1.0f + __expf(-zi));
      float gf = 1.0f / (1.0f + __expf(-zf));
      float gg = tanhf(zg);
      float go = 1.0f / (1.0f + __expf(-zo));
      float c  = fmaf(gf, cbuf[r * (D_ + 1) + j], gi * gg);
      cbuf[r * (D_ + 1) + j] = c;
      float h = go * tanhf(c);
      abuf[r * KP + INPAD + j] = h;
      hout[((size_t)(b0 + r) * T_ + tt) * (2 * D_) + dir * D_ + j] = h;
    }
    __syncthreads();
  }
}

// relu(h2 @ w_d1^T + b1) -> relu(@ w_d2^T + b2) -> @ w_out^T + b_out
__global__ void head_kernel(const float* __restrict__ h2,
                            const float* __restrict__ w_d1, const float* __restrict__ b_d1,
                            const float* __restrict__ w_d2, const float* __restrict__ b_d2,
                            const float* __restrict__ w_o,  const float* __restrict__ b_o,
                            float* __restrict__ out)
{
  __shared__ float w1[H1_ * IN2_];
  __shared__ float b1s[H1_];
  __shared__ float w2[H2_ * H1_];
  __shared__ float b2s[H2_];
  __shared__ float w3[H2_];
  __shared__ float b3s;
  const int tid = threadIdx.x;
  for (int i = tid; i < H1_ * IN2_; i += blockDim.x) w1[i] = w_d1[i];
  for (int i = tid; i < H2_ * H1_;  i += blockDim.x) w2[i] = w_d2[i];
  if (tid < H1_) b1s[tid] = b_d1[tid];
  if (tid < H2_) { b2s[tid] = b_d2[tid]; w3[tid] = w_o[tid]; }
  if (tid == 0) b3s = b_o[0];
  __syncthreads();

  const size_t row = (size_t)blockIdx.x * blockDim.x + tid;
  const float* hr = h2 + row * IN2_;
  float xv[IN2_];
  #pragma unroll
  for (int j = 0; j < IN2_; ++j) xv[j] = hr[j];

  float a1[H1_];
  for (int k = 0; k < H1_; ++k) {
    float s = b1s[k];
    #pragma unroll
    for (int j = 0; j < IN2_; ++j) s = fmaf(w1[k * IN2_ + j], xv[j], s);
    a1[k] = fmaxf(s, 0.0f);
  }
  float a2[H2_];
  for (int k = 0; k < H2_; ++k) {
    float s = b2s[k];
    #pragma unroll
    for (int j = 0; j < H1_; ++j) s = fmaf(w2[k * H1_ + j], a1[j], s);
    a2[k] = fmaxf(s, 0.0f);
  }
  float s = b3s;
  #pragma unroll
  for (int k = 0; k < H2_; ++k) s = fmaf(w3[k], a2[k], s);
  out[row] = s;
}

extern "C" void kernel_launch(void* const* d_in, const int* in_sizes, int n_in,
                              void* d_out, int out_size, void* d_ws, size_t ws_size,
                              hipStream_t stream) {
  (void)in_sizes; (void)n_in; (void)out_size; (void)ws_size;
  const float* x      = (const float*)d_in[0];
  const float* w_proj = (const float*)d_in[1];
  const float* b_proj = (const float*)d_in[2];
  const float* w_ih1f = (const float*)d_in[3];
  const float* w_hh1f = (const float*)d_in[4];
  const float* b_ih1f = (const float*)d_in[5];
  const float* b_hh1f = (const float*)d_in[6];
  const float* w_ih1b = (const float*)d_in[7];
  const float* w_hh1b = (const float*)d_in[8];
  const float* b_ih1b = (const float*)d_in[9];
  const float* b_hh1b = (const float*)d_in[10];
  const float* w_ih2f = (const float*)d_in[11];
  const float* w_hh2f = (const float*)d_in[12];
  const float* b_ih2f = (const float*)d_in[13];
  const float* b_hh2f = (const float*)d_in[14];
  const float* w_ih2b = (const float*)d_in[15];
  const float* w_hh2b = (const float*)d_in[16];
  const float* b_ih2b = (const float*)d_in[17];
  const float* b_hh2b = (const float*)d_in[18];
  const float* w_d1   = (const float*)d_in[19];
  const float* b_d1   = (const float*)d_in[20];
  const float* w_d2   = (const float*)d_in[21];
  const float* b_d2   = (const float*)d_in[22];
  const float* w_out  = (const float*)d_in[23];
  const float* b_out  = (const float*)d_in[24];

  // Workspace: h1 and h2 activation planes, fp32 [B,T,2D] each (~180.4 MB each).
  float* h1 = (float*)d_ws;
  float* h2 = h1 + (size_t)B_ * T_ * (2 * D_);

  // Layer 1: K = proj(43)->pad 44 | h(43)->pad 44 => KP=88.  Layer 2: 86->88 | 48 => KP=136.
  constexpr int KP1 = 88,  KW1 = 90;    // KW = KP + 2 (LDS bank spread, keeps 8B align)
  constexpr int KP2 = 136, KW2 = 138;
  constexpr size_t lds1 = (size_t)(GP_ * KW1 + GP_ + MT_ * GP_ + MT_ * KP1 +
                                   MT_ * (D_ + 1) + D_ * FEAT_ + D_) * sizeof(float);
  constexpr size_t lds2 = (size_t)(GP_ * KW2 + GP_ + MT_ * GP_ + MT_ * KP2 +
                                   MT_ * (D_ + 1) + D_ * FEAT_ + D_) * sizeof(float);

  const dim3 gridL((B_ / MT_) * 2);   // 16 batch tiles x 2 directions = 32 persistent WGs

  lstm_bidir_kernel<D_, 44, KP1, KW1, true><<<gridL, BLK_, lds1, stream>>>(
      x, w_proj, b_proj,
      w_ih1f, w_hh1f, b_ih1f, b_hh1f,
      w_ih1b, w_hh1b, b_ih1b, b_hh1b, h1);

  lstm_bidir_kernel<IN2_, 88, KP2, KW2, false><<<gridL, BLK_, lds2, stream>>>(
      h1, nullptr, nullptr,
      w_ih2f, w_hh2f, b_ih2f, b_hh2f,
      w_ih2b, w_hh2b, b_ih2b, b_hh2b, h2);

  head_kernel<<<(B_ * T_) / 256, 256, 0, stream>>>(
      h2, w_d1, b_d1, w_d2, b_d2, w_out, b_out, (float*)d_out);
}